// Coords2RMSD_35691178230505
// MI455X (gfx1250) — compile-verified
//
#include <hip/hip_runtime.h>
#include <hip/hip_bf16.h>

typedef __attribute__((ext_vector_type(2))) float v2f;
typedef __attribute__((ext_vector_type(8))) float v8f;
typedef __attribute__((ext_vector_type(4))) unsigned int v4u;
typedef __attribute__((ext_vector_type(8))) int v8i_t;
typedef __attribute__((ext_vector_type(4))) int v4i_t;

#define BATCH     8192
#define MATOMS    768
#define ROWLEN    (3 * MATOMS)          // floats per sample row (2304)
#define ROWBYTES  (ROWLEN * 4)          // 9216 B
#define SAMPLES_PER_WAVE 5
#define WAVES_PER_BLOCK  4
#define CHUNK_ATOMS      64             // atoms per TDM chunk
#define CHUNK_BYTES      (CHUNK_ATOMS * 12)   // 768 B per sample-row
#define NCHUNKS          (MATOMS / CHUNK_ATOMS)  // 12
#define REG_DW           1024           // dwords per LDS region (padded tile = 1005)
#define LDS_ROWSTRIDE_DW 201            // 192 data + 3*3 pad dwords per sample row

// LDS byte offset of a __shared__ object (strip the flat aperture via AS(3)).
__device__ __forceinline__ unsigned lds_byte_offset(const void* p) {
    return (unsigned)(unsigned long long)(__attribute__((address_space(3))) const char*)p;
}

// Issue one TDM 2-D tile load: nrows sample-rows of 192 dwords, global row
// stride 2304 dwords, LDS padded 3 dwords per 64 (bank-stagger, stride 201 dw).
__device__ __forceinline__ void tdm_load_2d(unsigned g_lo, unsigned g_hi,
                                            unsigned lds_addr, int nrows) {
    v4u g0;
    g0.x = 1u;                                        // count=1, user descriptor
    g0.y = lds_addr;                                  // LDS byte address
    g0.z = g_lo;                                      // global_addr[31:0]
    g0.w = (g_hi & 0x01FFFFFFu) | 0x80000000u;        // addr[56:32] | type=2
    v8i_t g1;
    g1[0] = 0x05520000;   // data_size=4B, pad_en, pad_interval=64dw, pad_amount=3dw
    g1[1] = (int)(192u << 16);                        // tensor_dim0[15:0]
    g1[2] = (int)(((unsigned)nrows) << 16);           // tensor_dim1[15:0]
    g1[3] = (int)(192u << 16);                        // tile_dim0 = 192 dwords
    g1[4] = nrows;                                    // tile_dim1 = rows
    g1[5] = ROWLEN;                                   // tensor_dim0_stride = 2304
    g1[6] = 0;
    g1[7] = 0;
    v4i_t z4 = {0, 0, 0, 0};
    v8i_t z8 = {0, 0, 0, 0, 0, 0, 0, 0};
    __builtin_amdgcn_tensor_load_to_lds(g0, g1, z4, z4, z8, 0);
}

// ---------------------------------------------------------------------------
// Kernel 1: one wave32 = 5 samples. TDM double-buffers 64-atom chunks of X/Y
// into LDS; WMMA f32 16x16x4 accumulates the batched 3xM * Mx3 correlation
// (block-diagonal 3x3 blocks of the 16x16 tile); VALU accumulates per-(s,comp)
// sums / sums-of-squares for the algebraic centering correction.
// ---------------------------------------------------------------------------
__global__ __launch_bounds__(128)
void kabsch_reduce_kernel(const float* __restrict__ X,
                          const float* __restrict__ Y,
                          const int*   __restrict__ num_atoms,
                          float* __restrict__ dump,   // [nwavesAlloc][32][8] f32
                          float* __restrict__ sums)   // [BATCH][12] f32
{
    __shared__ float lds[WAVES_PER_BLOCK * 4 * REG_DW];   // 64 KB

    const int lane  = threadIdx.x & 31;
    const int wave  = threadIdx.x >> 5;
    const int r     = lane & 15;        // row of A / col of B in the 16x16 tile
    const int half  = lane >> 4;        // 0: K=0,1   1: K=2,3  (ISA A/B layout)
    const int waveGlobal = blockIdx.x * WAVES_PER_BLOCK + wave;
    const int base  = waveGlobal * SAMPLES_PER_WAVE;
    if (base >= BATCH) return;          // wave-uniform, no barriers used

    int nvalid = BATCH - base;
    if (nvalid > SAMPLES_PER_WAVE) nvalid = SAMPLES_PER_WAVE;
    nvalid = __builtin_amdgcn_readfirstlane(nvalid);

    const int  sl     = (r < 15) ? (r / 3) : 0;   // sample-in-wave
    const int  comp   = (r < 15) ? (r % 3) : 0;   // xyz component
    const int  sample = base + sl;
    const bool valid  = (r < 15) && (sample < BATCH);
    const int  sclamp = valid ? sample : (BATCH - 1);
    // Effective atom count: 0 for invalid lanes -> every per-atom mask false.
    const int  nEff   = valid ? num_atoms[sclamp] : 0;

    // LDS regions: [wave][buf*2 + matrix] * 1024 dwords
    const unsigned ldsWaveByte = __builtin_amdgcn_readfirstlane(
        lds_byte_offset(&lds[0]) + (unsigned)wave * (4u * REG_DW * 4u));
    const float* ldsWave = &lds[wave * 4 * REG_DW];
    const int rowbase = sl * LDS_ROWSTRIDE_DW;    // per-lane LDS row base (dw)

    const unsigned long long xg0 =
        (unsigned long long)X + (unsigned long long)base * ROWBYTES;
    const unsigned long long yg0 =
        (unsigned long long)Y + (unsigned long long)base * ROWBYTES;

    v8f acc = {};                       // 16x16 f32 accumulator (8 VGPRs)
    float sumX = 0.f, sumY = 0.f, ssqX = 0.f, ssqY = 0.f;

    // Prologue: chunk 0 -> buffer 0
    {
        unsigned lo = __builtin_amdgcn_readfirstlane((unsigned)xg0);
        unsigned hi = __builtin_amdgcn_readfirstlane((unsigned)(xg0 >> 32));
        tdm_load_2d(lo, hi, ldsWaveByte + 0u * (REG_DW * 4u), nvalid);
        lo = __builtin_amdgcn_readfirstlane((unsigned)yg0);
        hi = __builtin_amdgcn_readfirstlane((unsigned)(yg0 >> 32));
        tdm_load_2d(lo, hi, ldsWaveByte + 1u * (REG_DW * 4u), nvalid);
    }

    for (int c = 0; c < NCHUNKS; ++c) {
        const int buf = c & 1;
        if (c + 1 < NCHUNKS) {
            const unsigned long long off = (unsigned long long)(c + 1) * CHUNK_BYTES;
            const unsigned long long xg = xg0 + off, yg = yg0 + off;
            const unsigned nb = (unsigned)(buf ^ 1) * 2u;
            unsigned lo = __builtin_amdgcn_readfirstlane((unsigned)xg);
            unsigned hi = __builtin_amdgcn_readfirstlane((unsigned)(xg >> 32));
            tdm_load_2d(lo, hi, ldsWaveByte + (nb + 0u) * (REG_DW * 4u), nvalid);
            lo = __builtin_amdgcn_readfirstlane((unsigned)yg);
            hi = __builtin_amdgcn_readfirstlane((unsigned)(yg >> 32));
            tdm_load_2d(lo, hi, ldsWaveByte + (nb + 1u) * (REG_DW * 4u), nvalid);
            __builtin_amdgcn_s_wait_tensorcnt(2);   // chunk c has landed
        } else {
            __builtin_amdgcn_s_wait_tensorcnt(0);
        }
        asm volatile("" ::: "memory");              // fence LDS reads vs TDM

        const float* xb = ldsWave + (buf * 2 + 0) * REG_DW;
        const float* yb = ldsWave + (buf * 2 + 1) * REG_DW;
        const int abase = c * CHUNK_ATOMS;

#pragma unroll 4
        for (int ab = 0; ab < CHUNK_ATOMS; ab += 4) {
            const int a0 = ab + 2 * half;           // K index per ISA A/B layout
            const int a1 = a0 + 1;
            const int i0 = 3 * a0 + comp;
            const int i1 = i0 + 3;
            const int d0 = rowbase + i0 + 3 * (i0 >> 6);
            const int d1 = rowbase + i1 + 3 * (i1 >> 6);

            // Unconditional LDS loads (always in-bounds) ...
            float ax = xb[d0];
            float ay = xb[d1];
            float bx = yb[d0];
            float by = yb[d1];
            // ... then flat v_cndmask selects (no EXEC manipulation).
            const bool k0 = (abase + a0) < nEff;
            const bool k1 = (abase + a1) < nEff;
            ax = k0 ? ax : 0.0f;
            ay = k1 ? ay : 0.0f;
            bx = k0 ? bx : 0.0f;
            by = k1 ? by : 0.0f;

            sumX += ax + ay;            sumY += bx + by;
            ssqX += ax * ax + ay * ay;  ssqY += bx * bx + by * by;

            v2f av = {ax, ay};
            v2f bv = {bx, by};
            acc = __builtin_amdgcn_wmma_f32_16x16x4_f32(
                      false, av, false, bv, (short)0, acc, false, false);
        }
    }

    // Combine the two K-halves of the per-lane scalar sums.
    sumX += __shfl_xor(sumX, 16);
    sumY += __shfl_xor(sumY, 16);
    ssqX += __shfl_xor(ssqX, 16);
    ssqY += __shfl_xor(ssqY, 16);

    if (valid && half == 0) {
        float* sw = sums + (size_t)sample * 12;
        sw[comp]     = sumX;
        sw[3 + comp] = sumY;
        sw[6 + comp] = ssqX;
        sw[9 + comp] = ssqY;
    }

    // Dump the accumulator tile: lane l = (M>=8)<<4 | N holds rows M%8 of col N.
    v8f* dst = (v8f*)dump + (size_t)waveGlobal * 32 + lane;
    *dst = acc;
}

// ---------------------------------------------------------------------------
// Kernel 2: 1 thread/sample. Rebuild centered R, singular values via the
// trigonometric eigen-solver of R^T R, det-sign correction, final RMSD.
// ---------------------------------------------------------------------------
__global__ __launch_bounds__(256)
void kabsch_finalize_kernel(const float* __restrict__ dump,
                            const float* __restrict__ sums,
                            const int*   __restrict__ num_atoms,
                            float* __restrict__ out)
{
    const int s = blockIdx.x * blockDim.x + threadIdx.x;
    if (s >= BATCH) return;

    const float n = (float)num_atoms[s];
    const float inv_n = 1.0f / n;
    const int g = s / SAMPLES_PER_WAVE;
    const int sloc = s - g * SAMPLES_PER_WAVE;
    const float* dw = dump + (size_t)g * 256;
    const float* sw = sums + (size_t)s * 12;

    float mx[3], my[3];
    float ssx = 0.f, ssy = 0.f;
#pragma unroll
    for (int i = 0; i < 3; ++i) {
        mx[i] = sw[i]     * inv_n;
        my[i] = sw[3 + i] * inv_n;
        ssx  += sw[6 + i];
        ssy  += sw[9 + i];
    }

    float R[3][3];
#pragma unroll
    for (int i = 0; i < 3; ++i) {
#pragma unroll
        for (int j = 0; j < 3; ++j) {
            const int m = 3 * sloc + i;           // row of tile
            const int c = 3 * sloc + j;           // col of tile
            const int ln = ((m >> 3) << 4) | c;   // lane that owns it
            const float sxy = dw[ln * 8 + (m & 7)];
            R[i][j] = sxy - n * mx[i] * my[j];    // centered correlation
        }
    }

    const float ex = ssx - n * (mx[0]*mx[0] + mx[1]*mx[1] + mx[2]*mx[2]);
    const float ey = ssy - n * (my[0]*my[0] + my[1]*my[1] + my[2]*my[2]);

    const float det =
        R[0][0] * (R[1][1]*R[2][2] - R[1][2]*R[2][1]) -
        R[0][1] * (R[1][0]*R[2][2] - R[1][2]*R[2][0]) +
        R[0][2] * (R[1][0]*R[2][1] - R[1][1]*R[2][0]);
    const float d = (det > 0.f) ? 1.f : ((det < 0.f) ? -1.f : 0.f);

    // A = R^T R (symmetric PSD)
    const float A00 = R[0][0]*R[0][0] + R[1][0]*R[1][0] + R[2][0]*R[2][0];
    const float A11 = R[0][1]*R[0][1] + R[1][1]*R[1][1] + R[2][1]*R[2][1];
    const float A22 = R[0][2]*R[0][2] + R[1][2]*R[1][2] + R[2][2]*R[2][2];
    const float A01 = R[0][0]*R[0][1] + R[1][0]*R[1][1] + R[2][0]*R[2][1];
    const float A02 = R[0][0]*R[0][2] + R[1][0]*R[1][2] + R[2][0]*R[2][2];
    const float A12 = R[0][1]*R[0][2] + R[1][1]*R[1][2] + R[2][1]*R[2][2];

    const float q  = (A00 + A11 + A22) * (1.0f / 3.0f);
    const float p1 = A01*A01 + A02*A02 + A12*A12;
    const float b00 = A00 - q, b11 = A11 - q, b22 = A22 - q;
    const float p2 = b00*b00 + b11*b11 + b22*b22 + 2.0f * p1;

    float e1, e2, e3;
    if (p2 <= 1e-24f) {
        e1 = e2 = e3 = q;
    } else {
        const float p = sqrtf(p2 * (1.0f / 6.0f));
        const float ip = 1.0f / p;
        const float B00 = b00*ip, B11 = b11*ip, B22 = b22*ip;
        const float B01 = A01*ip, B02 = A02*ip, B12 = A12*ip;
        float detB = B00*(B11*B22 - B12*B12)
                   - B01*(B01*B22 - B12*B02)
                   + B02*(B01*B12 - B11*B02);
        float rr = fminf(1.0f, fmaxf(-1.0f, 0.5f * detB));
        const float phi = acosf(rr) * (1.0f / 3.0f);
        e1 = q + 2.0f * p * cosf(phi);
        e3 = q + 2.0f * p * cosf(phi + 2.09439510239319549f); // + 2*pi/3
        e2 = 3.0f * q - e1 - e3;
    }

    const float s0 = sqrtf(fmaxf(e1, 0.f));
    const float s1 = sqrtf(fmaxf(e2, 0.f));
    const float s2 = sqrtf(fmaxf(e3, 0.f));
    const float tr = s0 + s1 + d * s2;
    const float e  = ex + ey - 2.0f * tr;
    out[s] = sqrtf(fmaxf(e, 0.f) * inv_n + 1e-7f);
}

// ---------------------------------------------------------------------------
extern "C" void kernel_launch(void* const* d_in, const int* in_sizes, int n_in,
                              void* d_out, int out_size, void* d_ws, size_t ws_size,
                              hipStream_t stream) {
    const float* X  = (const float*)d_in[0];
    const float* Y  = (const float*)d_in[1];
    const int*   na = (const int*)d_in[2];
    float* out = (float*)d_out;
    float* ws  = (float*)d_ws;

    const int nwaves  = (BATCH + SAMPLES_PER_WAVE - 1) / SAMPLES_PER_WAVE; // 1639
    const int blocks1 = (nwaves + WAVES_PER_BLOCK - 1) / WAVES_PER_BLOCK;  // 410
    const int nwavesAlloc = blocks1 * WAVES_PER_BLOCK;                     // 1640

    float* dump = ws;                                        // nwavesAlloc*256 f32
    float* sums = ws + (size_t)nwavesAlloc * 256;            // BATCH*12 f32
    // total scratch ~2.1 MB

    kabsch_reduce_kernel<<<blocks1, 128, 0, stream>>>(X, Y, na, dump, sums);
    kabsch_finalize_kernel<<<(BATCH + 255) / 256, 256, 0, stream>>>(dump, sums, na, out);
}